// SpikingLayer_17721035064087
// MI455X (gfx1250) — compile-verified
//
#include <hip/hip_runtime.h>

// ---------------------------------------------------------------------------
// Fused SNN layer for MI455X (gfx1250, wave32):
//   inputs = X @ W^T  via V_WMMA_F32_16X16X4_F32  (fused, never hits HBM)
//   LIF scan over T=500 carried in registers, 1 thread = 1 hidden unit
//   X tile staged via async-to-LDS when available; bank-conflict-free pitches
// ---------------------------------------------------------------------------

typedef __attribute__((ext_vector_type(2))) float v2f;
typedef __attribute__((ext_vector_type(4))) float v4f;
typedef __attribute__((ext_vector_type(8))) float v8f;
typedef __attribute__((ext_vector_type(4))) int   v4i;

#define B_   128
#define T_   500
#define C_   700
#define H_   1024
#define HT   128          // hidden units per workgroup
#define TB   16           // time steps per block (= WMMA M)
#define LXP  708          // X-tile pitch: 708 % 64 == 4 -> conflict-free A-frag reads
#define LIP  132          // inp-tile pitch: hi-half rows shift 32 banks -> conflict-free

// exp(-DT/TAU_RISE), exp(-DT/TAU_SYN), exp(-DT/TAU_MEM)
#define PHI_   0.36787944117144233f
#define GAMMA_ 0.8187307530779818f
#define BETA_  0.9048374180359595f

#if defined(__has_builtin)
#if __has_builtin(__builtin_amdgcn_global_load_async_to_lds_b128) && \
    __has_builtin(__builtin_amdgcn_s_wait_asynccnt)
#define USE_ASYNC_LDS 1
#endif
#endif

#ifdef USE_ASYNC_LDS
// Builtin signature (from hipcc diagnostic): (int4 AS1*, int4 AS3*, imm, imm)
typedef __attribute__((address_space(1))) v4i gv4i_t;   // global int4
typedef __attribute__((address_space(3))) v4i lv4i_t;   // LDS int4
#endif

__global__ __launch_bounds__(HT)
void snn_fused_kernel(const float* __restrict__ X,
                      const float* __restrict__ W,
                      float* __restrict__ U,
                      float* __restrict__ S)
{
    __shared__ float lX[TB * LXP];    // 16 x 708 fp32 = 45312 B
    __shared__ float lInp[TB * LIP];  // 16 x 132 fp32 =  8448 B

    const int b    = blockIdx.y;
    const int h0   = blockIdx.x * HT;
    const int tid  = threadIdx.x;
    const int lane = tid & 31;
    const int wave = tid >> 5;

    // --- per-thread scan state for hidden unit h = h0 + tid ---
    float sH = 0.f, sI = 0.f, sU = 0.f, sS = 0.f;
    const int h = h0 + tid;

    // t = 0 outputs (U_REST = 0, no spike)
    {
        size_t o0 = ((size_t)b * T_ + 0) * H_ + h;
        U[o0] = 0.f;
        S[o0] = 0.f;
    }

    // WMMA fragment lane mapping (ISA 7.12.2, 32-bit 16x4 A / 4x16 B):
    //   A: lanes 0-15 hold row M=lane, K=0..1 in v0,v1; lanes 16-31 hold K=2..3
    //   B: lane half selects K pair, lane&15 selects column N
    const int arow  = lane & 15;          // M (time row within tile)
    const int akoff = (lane >> 4) * 2;    // K sub-offset {0,2}
    const int nIdx  = lane & 15;          // N (hidden col within tile)

    const int NBLK = (T_ - 1 + TB - 1) / TB;   // 32 blocks cover inputs t=0..498

    for (int j = 0; j < NBLK; ++j) {
        const int tin0 = j * TB;

        // ---- 1) stage X[b, tin0:tin0+16, 0:700] into LDS (zero-fill OOB) ----
        for (int i = tid; i < TB * (C_ / 4); i += HT) {
            int r   = i / (C_ / 4);
            int c4  = i - r * (C_ / 4);
            int tin = tin0 + r;
            float* ldst = &lX[r * LXP + c4 * 4];
            if (tin < T_) {
                const float* gsrc = X + ((size_t)b * T_ + tin) * (size_t)C_ + c4 * 4;
#ifdef USE_ASYNC_LDS
                // dedicated async path: no VGPR round-trip, tracked by ASYNCcnt
                __builtin_amdgcn_global_load_async_to_lds_b128(
                    (gv4i_t*)gsrc, (lv4i_t*)ldst, 0, 0);
#else
                *(v4f*)ldst = *(const v4f*)gsrc;
#endif
            } else {
                v4f z = {0.f, 0.f, 0.f, 0.f};
                *(v4f*)ldst = z;
            }
        }
#ifdef USE_ASYNC_LDS
        __builtin_amdgcn_s_wait_asynccnt(0);
#endif
        __syncthreads();

        // ---- 2) GEMM: wave w computes tiles n0 = (2w+tt)*16  (K = 700) ----
        for (int tt = 0; tt < 2; ++tt) {
            const int n0 = (2 * wave + tt) * 16;
            const float* wrow = W + (size_t)(h0 + n0 + nIdx) * C_ + akoff;
            const float* ap   = &lX[arow * LXP + akoff];

            v8f acc = {0.f, 0.f, 0.f, 0.f, 0.f, 0.f, 0.f, 0.f};
#pragma unroll 5
            for (int k = 0; k < C_; k += 4) {
                v2f a  = *(const v2f*)(ap + k);        // ds_load_b64, conflict-free
                v2f bm = *(const v2f*)(wrow + k);      // global_load_b64 (L2-hot W)
                acc = __builtin_amdgcn_wmma_f32_16x16x4_f32(
                          false, a, false, bm, (short)0, acc, false, false);
            }
            // D layout: lane<16 -> M=r, lane>=16 -> M=r+8; N = lane&15
            const int mbase = (lane >> 4) * 8;
#pragma unroll
            for (int r = 0; r < 8; ++r)
                lInp[(mbase + r) * LIP + n0 + nIdx] = acc[r];
        }
        __syncthreads();

        // ---- 3) LIF scan: 16 sequential steps, state in registers ----
#pragma unroll
        for (int i = 0; i < TB; ++i) {
            float inp_prev = lInp[i * LIP + tid];      // inputs[t-1], conflict-free
            float Hc = PHI_ * sH + inp_prev;
            float Ic = GAMMA_ * sI + sH;
            float Uc = BETA_ * (sU - sI) - sS;         // R_MEM=1, U_REST=0, U_TH=1
            float Sc = (Uc > 1.0f) ? 1.0f : 0.0f;
            int t = tin0 + i + 1;
            if (t < T_) {
                size_t o = ((size_t)b * T_ + t) * H_ + h;
                U[o] = Uc;                              // coalesced 128B/wave
                S[o] = Sc;
            }
            sH = Hc; sI = Ic; sU = Uc; sS = Sc;
        }
        __syncthreads();
    }
}

extern "C" void kernel_launch(void* const* d_in, const int* in_sizes, int n_in,
                              void* d_out, int out_size, void* d_ws, size_t ws_size,
                              hipStream_t stream)
{
    (void)in_sizes; (void)n_in; (void)d_ws; (void)ws_size; (void)out_size;

    const float* X = (const float*)d_in[0];   // [128, 500, 700] fp32
    const float* W = (const float*)d_in[1];   // [1024, 700]     fp32

    float* U = (float*)d_out;                                   // [128,500,1024]
    float* S = (float*)d_out + (size_t)B_ * T_ * H_;            // [128,500,1024]

    dim3 grid(H_ / HT, B_);   // 8 x 128 = 1024 workgroups
    dim3 block(HT);           // 128 threads = 4 wave32

    snn_fused_kernel<<<grid, block, 0, stream>>>(X, W, U, S);
}